// GNN_39092792328217
// MI455X (gfx1250) — compile-verified
//
#include <hip/hip_runtime.h>
#include <cstddef>

#define D_FEAT 48
#define WPB 8   // waves per block in gcn_layer (256 threads)

#if defined(__AMDGCN__) && __has_builtin(__builtin_amdgcn_global_load_async_to_lds_b32) && __has_builtin(__builtin_amdgcn_s_wait_asynccnt)
#define USE_ASYNC_LDS 1
typedef __attribute__((address_space(1))) int as1_int;
typedef __attribute__((address_space(3))) int as3_int;
#else
#define USE_ASYNC_LDS 0
#endif

// ---------------------------------------------------------------- zero deg
__global__ void gcn_zero_i32(int* __restrict__ a, int n) {
    int i = blockIdx.x * blockDim.x + threadIdx.x;
    if (i < n) a[i] = 0;
}

// ---------------------------------------------------------------- in-degree count (dest = row)
__global__ void gcn_count_deg(const int* __restrict__ row, int* __restrict__ deg, int e) {
    int i = blockIdx.x * blockDim.x + threadIdx.x;
    if (i < e) atomicAdd(&deg[row[i]], 1);
}

// ---------------------------------------------------------------- single-block scan:
// off = exclusive_prefix(deg), off[n] = total, dinv = rsqrt(deg+1), cnt = 0
__global__ __launch_bounds__(1024) void gcn_scan(const int* __restrict__ deg,
                                                 int* __restrict__ off,
                                                 float* __restrict__ dinv,
                                                 int* __restrict__ cnt, int n) {
    __shared__ int sdata[1024];
    __shared__ int scarry;
    int tid = threadIdx.x;
    if (tid == 0) scarry = 0;
    __syncthreads();
    for (int base = 0; base < n; base += 1024) {
        int i = base + tid;
        int v = (i < n) ? deg[i] : 0;
        sdata[tid] = v;
        __syncthreads();
        // Hillis-Steele inclusive scan
        for (int s = 1; s < 1024; s <<= 1) {
            int t = (tid >= s) ? sdata[tid - s] : 0;
            __syncthreads();
            sdata[tid] += t;
            __syncthreads();
        }
        int incl  = sdata[tid];
        int carry = scarry;               // read before update barrier
        if (i < n) {
            off[i]  = carry + (incl - v); // exclusive
            dinv[i] = rsqrtf((float)(v + 1)); // +1 self loop; always > 0
            cnt[i]  = 0;
        }
        __syncthreads();
        if (tid == 1023) scarry = carry + incl;
        __syncthreads();
    }
    if (tid == 0) off[n] = scarry;
}

// ---------------------------------------------------------------- CSR fill (bucket by dest row)
__global__ void gcn_fill(const int* __restrict__ row, const int* __restrict__ col,
                         const int* __restrict__ off, int* __restrict__ cnt,
                         int* __restrict__ csr_col, int e) {
    int i = blockIdx.x * blockDim.x + threadIdx.x;
    if (i < e) {
        int r = row[i];
        int p = off[r] + atomicAdd(&cnt[r], 1);
        csr_col[p] = col[i];
    }
}

// ---------------------------------------------------------------- gather SpMM layer:
// one wave32 per node; lane l holds feat l and feat 32+(l&15); no atomics.
// gfx1250: neighbor-index chunks are double-buffered LDS stages filled by
// global_load_async_to_lds_b32 and synchronized with s_wait_asynccnt.
__global__ __launch_bounds__(256) void gcn_layer(const int* __restrict__ off,
                                                 const int* __restrict__ csr_col,
                                                 const float* __restrict__ dinv,
                                                 const float* __restrict__ xin,
                                                 float* __restrict__ xout, int n) {
#if USE_ASYNC_LDS
    __shared__ __align__(16) int stage[WPB][2][32];
#endif
    const int lane = threadIdx.x & 31;
    const int wv_  = threadIdx.x >> 5;
    const int node = blockIdx.x * WPB + wv_;
    if (node >= n) return;                    // uniform per wave

    const int idx1 = 32 + (lane & 15);        // lanes 16-31 duplicate low half (same cacheline)
    const float di = dinv[node];
    const float wself = di * di;              // self-loop weight dinv[i]^2

    const float* __restrict__ xself = xin + (size_t)node * D_FEAT;
    float acc0 = wself * xself[lane];
    float acc1 = wself * xself[idx1];

    const int s = off[node];
    const int e = off[node + 1];

#if USE_ASYNC_LDS
    // prologue: async-stage first chunk (per-lane 4B, 128B total per wave)
    if (s + lane < e) {
        __builtin_amdgcn_global_load_async_to_lds_b32(
            (as1_int*)(csr_col + s + lane),
            (as3_int*)&stage[wv_][0][lane], 0, 0);
    }
    int par = 0;
    for (int base = s; base < e; base += 32) {
        if (base + 32 < e) {
            // issue next chunk into the other buffer, then wait for current one
            const int jn = base + 32 + lane;
            if (jn < e) {
                __builtin_amdgcn_global_load_async_to_lds_b32(
                    (as1_int*)(csr_col + jn),
                    (as3_int*)&stage[wv_][par ^ 1][lane], 0, 0);
            }
            __builtin_amdgcn_s_wait_asynccnt(1);  // in-order: current chunk done
        } else {
            __builtin_amdgcn_s_wait_asynccnt(0);
        }
        const int j  = base + lane;
        const int c  = (j < e) ? stage[wv_][par][lane] : 0;
        const float wv = (j < e) ? di * dinv[c] : 0.0f;
        const int m = min(32, e - base);
        for (int k = 0; k < m; ++k) {
            const int   cc = __shfl(c,  k, 32);  // broadcast neighbor id
            const float ww = __shfl(wv, k, 32);  // broadcast edge weight
            const float* __restrict__ src = xin + (size_t)cc * D_FEAT;
            acc0 = fmaf(ww, src[lane], acc0);    // 128B coalesced L2 read
            acc1 = fmaf(ww, src[idx1], acc1);
        }
        par ^= 1;
    }
#else
    for (int base = s; base < e; base += 32) {
        const int j = base + lane;
        const int inb = (j < e);
        const int   c  = inb ? csr_col[j] : 0;
        const float wv = inb ? di * dinv[c] : 0.0f;
        if (base + 32 < e) __builtin_prefetch(&csr_col[base + 32 + lane], 0, 0);
        const int m = min(32, e - base);
        for (int k = 0; k < m; ++k) {
            const int   cc = __shfl(c,  k, 32);
            const float ww = __shfl(wv, k, 32);
            const float* __restrict__ src = xin + (size_t)cc * D_FEAT;
            acc0 = fmaf(ww, src[lane], acc0);
            acc1 = fmaf(ww, src[idx1], acc1);
        }
    }
#endif

    float* __restrict__ dst = xout + (size_t)node * D_FEAT;
    dst[lane] = acc0;
    if (lane < 16) dst[32 + lane] = acc1;
}

// ---------------------------------------------------------------- launch
extern "C" void kernel_launch(void* const* d_in, const int* in_sizes, int n_in,
                              void* d_out, int out_size, void* d_ws, size_t ws_size,
                              hipStream_t stream) {
    const int*   row = (const int*)d_in[0];
    const int*   col = (const int*)d_in[1];
    const float* x   = (const float*)d_in[2];
    float*       out = (float*)d_out;

    const int E = in_sizes[0];
    const int N = in_sizes[2] / D_FEAT;

    // carve scratch (256B-aligned regions)
    char* p = (char*)d_ws;
    auto take = [&](size_t bytes) -> char* {
        char* r = p;
        p += (bytes + 255) & ~(size_t)255;
        return r;
    };
    int*   deg  = (int*)  take((size_t)N * 4);
    int*   cnt  = (int*)  take((size_t)N * 4);
    int*   off  = (int*)  take((size_t)(N + 1) * 4);
    int*   csr  = (int*)  take((size_t)E * 4 + 128);   // +pad
    float* dinv = (float*)take((size_t)N * 4);
    float* xtmp = (float*)take((size_t)N * D_FEAT * 4);
    (void)ws_size; (void)n_in; (void)out_size;

    gcn_zero_i32 <<<(N + 255) / 256, 256, 0, stream>>>(deg, N);
    gcn_count_deg<<<(E + 255) / 256, 256, 0, stream>>>(row, deg, E);
    gcn_scan     <<<1, 1024, 0, stream>>>(deg, off, dinv, cnt, N);
    gcn_fill     <<<(E + 255) / 256, 256, 0, stream>>>(row, col, off, cnt, csr, E);

    const int nBlocks = (N + WPB - 1) / WPB;
    gcn_layer<<<nBlocks, 256, 0, stream>>>(off, csr, dinv, x,    xtmp, N);
    gcn_layer<<<nBlocks, 256, 0, stream>>>(off, csr, dinv, xtmp, out,  N);
}